// MST_ZRD_3118146257294
// MI455X (gfx1250) — compile-verified
//
#include <hip/hip_runtime.h>
#include <cstddef>

typedef float v2f __attribute__((ext_vector_type(2)));
typedef float v8f __attribute__((ext_vector_type(8)));

#define NC   28
#define NCP  32                     // padded channel count (zero rows 28..31)
#define IMG  256
#define HW   (IMG * IMG)            // 65536
#define CHW  (NC * HW)              // 1835008
#define COLX 310                    // 256 + 27*2
#define NPIX 400                    // 20x20 halo tile
#define NB   16                     // batch

// LDS layout (floats)
#define OFF_M1   0                          // [32][400]
#define OFF_T    (OFF_M1 + NCP * NPIX)      // [32][400]
#define OFF_W1   (OFF_T  + NCP * NPIX)      // [32][28]
#define OFF_W2   (OFF_W1 + NCP * NC)
#define OFF_B1   (OFF_W2 + NCP * NC)        // [32]
#define OFF_B2   (OFF_B1 + NCP)
#define OFF_DW   (OFF_B2 + NCP)             // [28][25]
#define OFF_DB   (OFF_DW + NC * 25)         // [28]
#define LDS_FLOATS (OFF_DB + NC)

static __device__ __forceinline__ v8f wmma4(v2f a, v2f b, v8f c) {
    // D(16x16,f32) = A(16x4,f32) * B(4x16,f32) + C
    return __builtin_amdgcn_wmma_f32_16x16x4_f32(
        /*neg_a=*/false, a, /*neg_b=*/false, b,
        /*c_mod=*/(short)0, c, /*reuse_a=*/false, /*reuse_b=*/false);
}

// -----------------------------------------------------------------------------
// Kernel 1: fused mask_guided for both parameter sets.
// grid = (16 tilesX, 16 tilesY, 16 batch), block = 256 threads (8 waves).
// -----------------------------------------------------------------------------
__global__ void k_mask_guided(
    const float* __restrict__ Phi,
    const float* __restrict__ w1m, const float* __restrict__ b1m,
    const float* __restrict__ w2m, const float* __restrict__ b2m,
    const float* __restrict__ dwm, const float* __restrict__ dbm,
    const float* __restrict__ w1p, const float* __restrict__ b1p,
    const float* __restrict__ w2p, const float* __restrict__ b2p,
    const float* __restrict__ dwp, const float* __restrict__ dbp,
    float* __restrict__ Pmm, float* __restrict__ Ppv)
{
    extern __shared__ float smem[];
    float* m1L = smem + OFF_M1;     // [ch][p], p contiguous
    float* tL  = smem + OFF_T;
    float* w1s = smem + OFF_W1;     // [32][28], rows 28..31 zero
    float* w2s = smem + OFF_W2;
    float* b1s = smem + OFF_B1;     // [32], 28..31 zero
    float* b2s = smem + OFF_B2;
    float* dws = smem + OFF_DW;     // [28][25]
    float* dbs = smem + OFF_DB;     // [28]

    const int tx = blockIdx.x, ty = blockIdx.y, b = blockIdx.z;
    const int r0 = ty * 16 - 2;     // halo origin (may be negative)
    const int c0 = tx * 16 - 2;
    const int tid  = threadIdx.x;
    const int lane = tid & 31;
    const int wave = tid >> 5;
    const int mrow = lane & 15;     // M index (pixel in group) / N column
    const int kh   = lane >> 4;     // which K pair this lane holds

    const float* PhiB = Phi + (size_t)b * CHW;

    for (int br = 0; br < 2; ++br) {
        const float* w1 = br ? w1p : w1m;
        const float* b1 = br ? b1p : b1m;
        const float* w2 = br ? w2p : w2m;
        const float* b2 = br ? b2p : b2m;
        const float* dw = br ? dwp : dwm;
        const float* db = br ? dbp : dbm;
        float*       P  = br ? Ppv : Pmm;

        // ---- stage padded weights / biases / dw in LDS ----------------------
        for (int i = tid; i < NCP * NC; i += 256) {      // 896
            const float v = (i < NC * NC) ? w1[i] : 0.0f;
            const float u = (i < NC * NC) ? w2[i] : 0.0f;
            w1s[i] = v;
            w2s[i] = u;
        }
        for (int i = tid; i < NC * 25; i += 256) dws[i] = dw[i];  // 700
        if (tid < NCP) {
            b1s[tid] = (tid < NC) ? b1[tid] : 0.0f;
            b2s[tid] = (tid < NC) ? b2[tid] : 0.0f;
            if (tid < NC) dbs[tid] = db[tid];
        }
        __syncthreads();

        // ---------------- GEMM1: m1 = Phi @ W1^T + b1 over 400 halo pixels ---
        for (int g = wave; g < 25; g += 8) {
            const int p  = g * 16 + mrow;           // pixel index in halo tile
            const int rr = p / 20, cc = p % 20;
            const int gr = r0 + rr, gc = c0 + cc;
            const bool inb = (gr >= 0) & (gr < IMG) & (gc >= 0) & (gc < IMG);
            const float msk = inb ? 1.0f : 0.0f;
            const int grc = min(max(gr, 0), IMG - 1);
            const int gcc = min(max(gc, 0), IMG - 1);
            const float* ap = PhiB + (size_t)grc * IMG + gcc;

            const int n  = mrow;                    // N column (output channel)
            const float bb0 = b1s[n];
            const float bb1 = b1s[n + 16];
            v8f acc0, acc1;
            #pragma unroll
            for (int j = 0; j < 8; ++j) { acc0[j] = bb0; acc1[j] = bb1; }

            #pragma unroll
            for (int kt = 0; kt < 7; ++kt) {
                const int k0 = kt * 4 + kh * 2;     // even, 0..26
                v2f a;
                a.x = ap[(size_t)k0 * HW] * msk;
                a.y = ap[(size_t)(k0 + 1) * HW] * msk;
                const v2f bA = *(const v2f*)&w1s[n * NC + k0];          // b64
                const v2f bB = *(const v2f*)&w1s[(n + 16) * NC + k0];   // b64
                acc0 = wmma4(a, bA, acc0);
                acc1 = wmma4(a, bB, acc1);
            }
            // D layout: vgpr r, lanes0-15 -> M=r ; lanes16-31 -> M=r+8
            const int pst = g * 16 + kh * 8;
            {
                float* d = &m1L[n * NPIX + pst];
                *(float4*)(d)     = make_float4(acc0[0], acc0[1], acc0[2], acc0[3]);
                *(float4*)(d + 4) = make_float4(acc0[4], acc0[5], acc0[6], acc0[7]);
            }
            {
                float* d = &m1L[(n + 16) * NPIX + pst];   // rows 28..31 scratch
                *(float4*)(d)     = make_float4(acc1[0], acc1[1], acc1[2], acc1[3]);
                *(float4*)(d + 4) = make_float4(acc1[4], acc1[5], acc1[6], acc1[7]);
            }
        }
        __syncthreads();

        // ---------------- GEMM2: t = m1 @ W2^T + b2 --------------------------
        for (int g = wave; g < 25; g += 8) {
            const int p  = g * 16 + mrow;
            const int n  = mrow;
            const float bb0 = b2s[n];
            const float bb1 = b2s[n + 16];
            v8f acc0, acc1;
            #pragma unroll
            for (int j = 0; j < 8; ++j) { acc0[j] = bb0; acc1[j] = bb1; }

            #pragma unroll
            for (int kt = 0; kt < 7; ++kt) {
                const int k0 = kt * 4 + kh * 2;
                v2f a;
                a.x = m1L[k0 * NPIX + p];
                a.y = m1L[(k0 + 1) * NPIX + p];
                const v2f bA = *(const v2f*)&w2s[n * NC + k0];
                const v2f bB = *(const v2f*)&w2s[(n + 16) * NC + k0];
                acc0 = wmma4(a, bA, acc0);
                acc1 = wmma4(a, bB, acc1);
            }
            const int pst = g * 16 + kh * 8;
            {
                float* d = &tL[n * NPIX + pst];
                *(float4*)(d)     = make_float4(acc0[0], acc0[1], acc0[2], acc0[3]);
                *(float4*)(d + 4) = make_float4(acc0[4], acc0[5], acc0[6], acc0[7]);
            }
            {
                float* d = &tL[(n + 16) * NPIX + pst];
                *(float4*)(d)     = make_float4(acc1[0], acc1[1], acc1[2], acc1[3]);
                *(float4*)(d + 4) = make_float4(acc1[4], acc1[5], acc1[6], acc1[7]);
            }
        }
        __syncthreads();

        // zero out-of-image halo pixels of t (matches SAME zero-padding)
        for (int p = tid; p < NPIX; p += 256) {
            const int rr = p / 20, cc = p % 20;
            const int gr = r0 + rr, gc = c0 + cc;
            if (gr < 0 || gr >= IMG || gc < 0 || gc >= IMG) {
                #pragma unroll
                for (int ch = 0; ch < NC; ++ch) tL[ch * NPIX + p] = 0.0f;
            }
        }
        __syncthreads();

        // depthwise 5x5 + bias -> sigmoid -> emb = m1*attn + m1 -> fixup -> P
        {
            const int ir = tid >> 4, ic = tid & 15;   // interior pixel
            const int gr = ty * 16 + ir, gc = tx * 16 + ic;
            const size_t obase = (size_t)b * CHW + (size_t)gr * IMG + gc;
            for (int ch = 0; ch < NC; ++ch) {
                float s = dbs[ch];
                #pragma unroll
                for (int dr = 0; dr < 5; ++dr) {
                    #pragma unroll
                    for (int dc = 0; dc < 5; ++dc) {
                        s = fmaf(dws[ch * 25 + dr * 5 + dc],
                                 tL[ch * NPIX + (ir + dr) * 20 + (ic + dc)], s);
                    }
                }
                const float attn = 1.0f / (1.0f + __expf(-s));
                const float m1v  = m1L[ch * NPIX + (ir + 2) * 20 + (ic + 2)];
                const float emb  = fmaf(m1v, attn, m1v);
                P[obase + (size_t)ch * HW] = (emb == 0.0f) ? 1e-6f : emb;
            }
        }
        __syncthreads();   // protect LDS before next branch reuses it
    }
}

// -----------------------------------------------------------------------------
// Kernel 2: temp = y - A_op(x_pre, Phi_learned).  One thread per [b,h,col].
// -----------------------------------------------------------------------------
__global__ void k_Aop(const float* __restrict__ y,
                      const float* __restrict__ xpre,
                      const float* __restrict__ Pmm,
                      float* __restrict__ temp)
{
    const int idx = blockIdx.x * 256 + threadIdx.x;
    const int total = NB * IMG * COLX;
    if (idx >= total) return;
    const int c  = idx % COLX;
    const int t1 = idx / COLX;
    const int h  = t1 % IMG;
    const int b  = t1 / IMG;

    const int ilo = (c > 255) ? ((c - 254) >> 1) : 0;   // ceil((c-255)/2)
    const int ihi = min(NC - 1, c >> 1);

    float s = 0.0f;
    for (int i = ilo; i <= ihi; ++i) {
        const size_t off = ((size_t)(b * NC + i) * IMG + h) * IMG + (c - 2 * i);
        s = fmaf(xpre[off], Pmm[off], s);
    }
    temp[idx] = y[idx] - s;
}

// -----------------------------------------------------------------------------
// Kernel 3: x_init = x_pre + delta * (temp * alpha windows) * Phi_pinv_learned
// -----------------------------------------------------------------------------
__global__ void k_final(const float* __restrict__ xpre,
                        const float* __restrict__ Ppv,
                        const float* __restrict__ temp,
                        const float* __restrict__ delta,
                        float* __restrict__ out)
{
    const int idx = blockIdx.x * 256 + threadIdx.x;
    const int total = NB * NC * IMG * IMG;
    if (idx >= total) return;
    const int w = idx & 255;
    int t = idx >> 8;
    const int h = t & 255;
    t >>= 8;
    const int i = t % NC;
    const int b = t / NC;

    const int col = 2 * i + w;
    const int jlo = (col > 255) ? ((col - 254) >> 1) : 0;
    const int jhi = min(NC - 1, col >> 1);
    const float alpha = 1.0f / (float)(jhi - jlo + 1);

    const float tv = temp[((size_t)b * IMG + h) * COLX + col];
    out[idx] = fmaf(delta[0], tv * alpha * Ppv[idx], xpre[idx]);
}

// -----------------------------------------------------------------------------
extern "C" void kernel_launch(void* const* d_in, const int* in_sizes, int n_in,
                              void* d_out, int out_size, void* d_ws, size_t ws_size,
                              hipStream_t stream)
{
    const float* y     = (const float*)d_in[0];
    const float* Phi   = (const float*)d_in[1];
    const float* xpre  = (const float*)d_in[2];
    const float* delta = (const float*)d_in[3];
    const float* mm_w1 = (const float*)d_in[4];
    const float* mm_b1 = (const float*)d_in[5];
    const float* mm_w2 = (const float*)d_in[6];
    const float* mm_b2 = (const float*)d_in[7];
    const float* mm_dw = (const float*)d_in[8];
    const float* mm_db = (const float*)d_in[9];
    const float* pv_w1 = (const float*)d_in[10];
    const float* pv_b1 = (const float*)d_in[11];
    const float* pv_w2 = (const float*)d_in[12];
    const float* pv_b2 = (const float*)d_in[13];
    const float* pv_dw = (const float*)d_in[14];
    const float* pv_db = (const float*)d_in[15];

    const size_t nMask = (size_t)NB * NC * HW;     // 29,360,128 floats
    float* Pmm  = (float*)d_ws;
    float* Ppv  = Pmm + nMask;
    float* temp = Ppv + nMask;

    const size_t ldsBytes = (size_t)LDS_FLOATS * sizeof(float);
    (void)hipFuncSetAttribute((const void*)k_mask_guided,
                              hipFuncAttributeMaxDynamicSharedMemorySize,
                              (int)ldsBytes);

    dim3 grid1(16, 16, 16);
    k_mask_guided<<<grid1, 256, ldsBytes, stream>>>(
        Phi,
        mm_w1, mm_b1, mm_w2, mm_b2, mm_dw, mm_db,
        pv_w1, pv_b1, pv_w2, pv_b2, pv_dw, pv_db,
        Pmm, Ppv);

    const int n2 = NB * IMG * COLX;
    k_Aop<<<(n2 + 255) / 256, 256, 0, stream>>>(y, xpre, Pmm, temp);

    const int n3 = NB * NC * IMG * IMG;
    k_final<<<(n3 + 255) / 256, 256, 0, stream>>>(xpre, Ppv, temp, delta,
                                                  (float*)d_out);
}